// DetectionTargetLayer_86517821216531
// MI455X (gfx1250) — compile-verified
//
#include <hip/hip_runtime.h>
#include <hip/hip_bf16.h>
#include <math.h>

// Problem constants (match reference)
#define N_IMG   4
#define N_PROP  2000
#define MAX_GT  64
#define P_POS   66      // int(200 * 0.33)
#define N_NEG   134     // 200 - 66
#define T_ROIS  200
#define MH      28
#define MW      28
#define IMG_H   512
#define IMG_W   512
#define EPSF    1e-8f

// d_out flat layout (floats): rois | class | deltas | masks
#define ROIS_OFF  0
#define CLASS_OFF 3200      // 4*200*4
#define DELTA_OFF 4000      // + 4*200
#define MASK_OFF  7200      // + 4*200*4

// ws layout (ints)
#define WS_POSF   0         // [4][2048]
#define WS_NEGF   8192      // [4][2048]
#define WS_BESTG  16384     // [4][2048]
#define WS_PIDX   24576     // [4][66]
#define WS_NIDX   25088     // [4][134]
#define WS_CNT    26112     // [4][2]

// Low 32 bits of a generic pointer to __shared__ = LDS byte offset on gfx1250
// (LDS aperture).
__device__ __forceinline__ unsigned lds_off_of(const void* p) {
  return (unsigned)(unsigned long long)p;
}

// ---------------------------------------------------------------------------
// Kernel A: per-proposal IoU stats. GT boxes/classes staged to LDS via the
// CDNA5 async global->LDS DMA path (ASYNCcnt-tracked).
// ---------------------------------------------------------------------------
__global__ void __launch_bounds__(256)
iou_classify(const float* __restrict__ proposals,
             const int*   __restrict__ class_ids,
             const float* __restrict__ gt_boxes,
             int* __restrict__ posf, int* __restrict__ negf,
             int* __restrict__ bestg)
{
  __shared__ float gtb[MAX_GT * 4];  // 1 KB
  __shared__ int   gcl[MAX_GT];      // 256 B
  const int img = blockIdx.y;
  const int tid = threadIdx.x;

  // Async-stage GT boxes: 64 lanes x 16B, and class ids: 16 lanes x 16B.
  {
    const float* gsrc = gt_boxes + (size_t)img * MAX_GT * 4;
    if (tid < MAX_GT) {
      unsigned ldst = lds_off_of(&gtb[tid * 4]);
      unsigned voff = (unsigned)tid * 16u;
      asm volatile("global_load_async_to_lds_b128 %0, %1, %2"
                   :: "v"(ldst), "v"(voff), "s"(gsrc) : "memory");
    }
    const int* csrc = class_ids + (size_t)img * MAX_GT;
    if (tid < MAX_GT / 4) {
      unsigned ldst = lds_off_of(&gcl[tid * 4]);
      unsigned voff = (unsigned)tid * 16u;
      asm volatile("global_load_async_to_lds_b128 %0, %1, %2"
                   :: "v"(ldst), "v"(voff), "s"(csrc) : "memory");
    }
    asm volatile("s_wait_asynccnt 0" ::: "memory");
  }
  __syncthreads();

  const int p = blockIdx.x * 256 + tid;
  if (p >= N_PROP) return;

  const float* pb = proposals + ((size_t)img * N_PROP + p) * 4;
  const float py1 = pb[0], px1 = pb[1], py2 = pb[2], px2 = pb[3];
  const bool pvalid = (py1 != 0.f) | (px1 != 0.f) | (py2 != 0.f) | (px2 != 0.f);
  const float a1 = (py2 - py1) * (px2 - px1);

  float ovmax = -1.0f, crmax = -1.0f;
  int best = 0;
#pragma unroll 4
  for (int g = 0; g < MAX_GT; ++g) {
    const float gy1 = gtb[g * 4 + 0], gx1 = gtb[g * 4 + 1];
    const float gy2 = gtb[g * 4 + 2], gx2 = gtb[g * 4 + 3];
    const int c = gcl[g];
    const bool bval = (gy1 != 0.f) | (gx1 != 0.f) | (gy2 != 0.f) | (gx2 != 0.f);
    const float ih = fmaxf(fminf(py2, gy2) - fmaxf(py1, gy1), 0.0f);
    const float iw = fmaxf(fminf(px2, gx2) - fmaxf(px1, gx1), 0.0f);
    const float inter = ih * iw;
    const float a2 = (gy2 - gy1) * (gx2 - gx1);
    const float iou = inter / (a1 + a2 - inter + EPSF);
    const float ovg = ((c > 0) && bval) ? iou : -1.0f;   // gt_valid mask
    if (ovg > ovmax) { ovmax = ovg; best = g; }          // first-max = argmax
    const float cg = ((c < 0) && bval) ? iou : -1.0f;    // crowd mask
    crmax = fmaxf(crmax, cg);
  }

  const bool pos = (ovmax >= 0.5f) && pvalid;
  const bool neg = (ovmax < 0.5f) && (crmax < 0.001f) && pvalid && !pos;
  const size_t o = (size_t)img * 2048 + p;
  posf[o] = pos ? 1 : 0;
  negf[o] = neg ? 1 : 0;
  bestg[o] = best;
}

// ---------------------------------------------------------------------------
// Kernel B: stable first-k compaction via block prefix scan (replaces the
// reference's 2000-element argsort). One block per image.
// ---------------------------------------------------------------------------
__global__ void __launch_bounds__(256)
select_rois(const int* __restrict__ posf, const int* __restrict__ negf,
            int* __restrict__ pidx, int* __restrict__ nidx,
            int* __restrict__ counts)
{
  __shared__ int s[256];
  __shared__ int sh_pc;
  const int img = blockIdx.x;
  const int t = threadIdx.x;
  const int base = img * 2048;
  const int i0 = t * 8;   // contiguous chunk of 8 -> stable index order

  // ---- positives ----
  int cnt = 0;
#pragma unroll
  for (int k = 0; k < 8; ++k) { int i = i0 + k; if (i < N_PROP) cnt += posf[base + i]; }
  s[t] = cnt; __syncthreads();
  for (int d = 1; d < 256; d <<= 1) {
    int v = (t >= d) ? s[t - d] : 0; __syncthreads();
    s[t] += v; __syncthreads();
  }
  int excl = s[t] - cnt;
  int total = s[255];
  int pc = total < P_POS ? total : P_POS;
  int rank = excl;
#pragma unroll
  for (int k = 0; k < 8; ++k) {
    int i = i0 + k;
    if (i < N_PROP && posf[base + i]) {
      if (rank < P_POS) pidx[img * P_POS + rank] = i;
      rank++;
    }
  }
  if (t == 0) { sh_pc = pc; counts[img * 2 + 0] = pc; }
  __syncthreads();

  // ---- negatives ----
  cnt = 0;
#pragma unroll
  for (int k = 0; k < 8; ++k) { int i = i0 + k; if (i < N_PROP) cnt += negf[base + i]; }
  s[t] = cnt; __syncthreads();
  for (int d = 1; d < 256; d <<= 1) {
    int v = (t >= d) ? s[t - d] : 0; __syncthreads();
    s[t] += v; __syncthreads();
  }
  excl = s[t] - cnt;
  int ntotal = s[255];
  pc = sh_pc;
  // exact reference arithmetic: int32(f32(pc) / f32(0.33)) - pc
  int neg_target = (int)((float)pc / 0.33f) - pc;
  int avail = ntotal < N_NEG ? ntotal : N_NEG;
  int nc = neg_target < 0 ? 0 : (neg_target > avail ? avail : neg_target);
  rank = excl;
#pragma unroll
  for (int k = 0; k < 8; ++k) {
    int i = i0 + k;
    if (i < N_PROP && negf[base + i]) {
      if (rank < N_NEG) nidx[img * N_NEG + rank] = i;
      rank++;
    }
  }
  if (t == 0) counts[img * 2 + 1] = nc;
}

// ---------------------------------------------------------------------------
// Kernel C: emit rois/class/deltas + 28x28 bilinear mask per roi slot.
// One block per (roi, image); scalar part is redundant-per-thread (cache hit),
// 784 mask pixels split across 256 lanes with coalesced stores.
// ---------------------------------------------------------------------------
__global__ void __launch_bounds__(256)
emit_targets(const float* __restrict__ proposals,
             const int*   __restrict__ class_ids,
             const float* __restrict__ gt_boxes,
             const unsigned char* __restrict__ masks,   // [4][512][512][64] bool
             const int* __restrict__ pidx, const int* __restrict__ nidx,
             const int* __restrict__ counts, const int* __restrict__ bestg,
             float* __restrict__ out)
{
  const int r   = blockIdx.x;   // roi slot 0..199
  const int img = blockIdx.y;
  const int tid = threadIdx.x;
  const int pc = counts[img * 2 + 0];
  const int nc = counts[img * 2 + 1];

  float b0 = 0.f, b1 = 0.f, b2 = 0.f, b3 = 0.f;
  float d0 = 0.f, d1 = 0.f, d2 = 0.f, d3 = 0.f;
  int cls = 0, gidx = 0;
  bool valid_pos = false;

  if (r < P_POS) {
    if (r < pc) {
      const int pi = pidx[img * P_POS + r];
      const float* pb = proposals + ((size_t)img * N_PROP + pi) * 4;
      b0 = pb[0]; b1 = pb[1]; b2 = pb[2]; b3 = pb[3];
      gidx = bestg[(size_t)img * 2048 + pi];
      cls = class_ids[img * MAX_GT + gidx];
      const float* gb = gt_boxes + ((size_t)img * MAX_GT + gidx) * 4;
      const float h  = fmaxf(b2 - b0, EPSF), w  = fmaxf(b3 - b1, EPSF);
      const float cy = b0 + 0.5f * h,        cx = b1 + 0.5f * w;
      const float gh = fmaxf(gb[2] - gb[0], EPSF), gw = fmaxf(gb[3] - gb[1], EPSF);
      const float gcy = gb[0] + 0.5f * gh,   gcx = gb[1] + 0.5f * gw;
      d0 = ((gcy - cy) / h) / 0.1f;
      d1 = ((gcx - cx) / w) / 0.1f;
      d2 = logf(gh / h) / 0.2f;
      d3 = logf(gw / w) / 0.2f;
      valid_pos = true;
    }
  } else {
    const int sneg = r - P_POS;
    if (sneg < nc) {
      const int ni = nidx[img * N_NEG + sneg];
      const float* pb = proposals + ((size_t)img * N_PROP + ni) * 4;
      b0 = pb[0]; b1 = pb[1]; b2 = pb[2]; b3 = pb[3];
    }
  }

  if (tid == 0) {
    float* ro = out + ROIS_OFF + ((size_t)img * T_ROIS + r) * 4;
    ro[0] = b0; ro[1] = b1; ro[2] = b2; ro[3] = b3;
    out[CLASS_OFF + img * T_ROIS + r] = (float)cls;
    float* dd = out + DELTA_OFF + ((size_t)img * T_ROIS + r) * 4;
    dd[0] = d0; dd[1] = d1; dd[2] = d2; dd[3] = d3;
  }

  float* mo = out + MASK_OFF + ((size_t)img * T_ROIS + r) * (MH * MW);
  for (int pix = tid; pix < MH * MW; pix += 256) {
    float v = 0.f;
    if (valid_pos) {
      const int py = pix / MW, px = pix % MW;
      const float fy = (float)py / 27.0f;
      const float fx = (float)px / 27.0f;
      const float ys = (b0 + fy * (b2 - b0)) * 511.0f;
      const float xs = (b1 + fx * (b3 - b1)) * 511.0f;
      const float y0f = floorf(ys), x0f = floorf(xs);
      const float ly = ys - y0f,    lx = xs - x0f;
      int y0 = (int)y0f; y0 = y0 < 0 ? 0 : (y0 > 511 ? 511 : y0);
      int y1 = y0 + 1;   y1 = y1 > 511 ? 511 : y1;
      int x0 = (int)x0f; x0 = x0 < 0 ? 0 : (x0 > 511 ? 511 : x0);
      int x1 = x0 + 1;   x1 = x1 > 511 ? 511 : x1;
      const size_t mb = (size_t)img * (512u * 512u * 64u) + (size_t)gidx;
      const float m00 = (float)masks[mb + ((size_t)y0 * 512 + x0) * 64];
      const float m01 = (float)masks[mb + ((size_t)y0 * 512 + x1) * 64];
      const float m10 = (float)masks[mb + ((size_t)y1 * 512 + x0) * 64];
      const float m11 = (float)masks[mb + ((size_t)y1 * 512 + x1) * 64];
      const float top = m00 * (1.f - lx) + m01 * lx;
      const float bot = m10 * (1.f - lx) + m11 * lx;
      v = rintf(top * (1.f - ly) + bot * ly);   // round-half-even == jnp.round
    }
    mo[pix] = v;
  }
}

// ---------------------------------------------------------------------------
extern "C" void kernel_launch(void* const* d_in, const int* in_sizes, int n_in,
                              void* d_out, int out_size, void* d_ws, size_t ws_size,
                              hipStream_t stream) {
  const float* proposals        = (const float*)d_in[0];  // [4][2000][4]
  const int*   class_ids        = (const int*)  d_in[1];  // [4][64]
  const float* gt_boxes         = (const float*)d_in[2];  // [4][64][4]
  const unsigned char* masks    = (const unsigned char*)d_in[3]; // [4][512][512][64] bool
  float* out = (float*)d_out;
  int* ws = (int*)d_ws;

  int* posf   = ws + WS_POSF;
  int* negf   = ws + WS_NEGF;
  int* bestg  = ws + WS_BESTG;
  int* pidx   = ws + WS_PIDX;
  int* nidx   = ws + WS_NIDX;
  int* counts = ws + WS_CNT;

  iou_classify<<<dim3((N_PROP + 255) / 256, N_IMG), 256, 0, stream>>>(
      proposals, class_ids, gt_boxes, posf, negf, bestg);
  select_rois<<<dim3(N_IMG), 256, 0, stream>>>(posf, negf, pidx, nidx, counts);
  emit_targets<<<dim3(T_ROIS, N_IMG), 256, 0, stream>>>(
      proposals, class_ids, gt_boxes, masks, pidx, nidx, counts, bestg, out);
}